// CriterionIFV_11527692222964
// MI455X (gfx1250) — compile-verified
//
#include <hip/hip_runtime.h>
#include <hip/hip_bf16.h>
#include <math.h>

// ---------------------------------------------------------------------------
// CriterionIFV on MI455X (gfx1250, wave32)
//
// Memory-bound problem: 2 passes over 302MB of fp32 features -> ~26us floor
// at 23.3 TB/s HBM. FLOPs negligible, so stay in fp32 throughout and use
// V_WMMA_F32_16X16X4_F32 for the pixel-vs-class-center dot products
// (GEMM: [pixels x 256ch] x [256ch x 19classes], classes padded to 32).
// ---------------------------------------------------------------------------

#define BB   8
#define CC   256
#define HH   96
#define WW   192
#define HWP  (HH * WW)            // 18432 pixels per batch image
#define NCLS 19
#define EPS_MEAN 1e-6f
#define EPS_COS  1e-8f
#define INV_N (1.0f / (float)(BB * HWP))

// workspace layout (floats)
#define OFF_SUMS_S  0
#define OFF_SUMS_T  (OFF_SUMS_S + BB * NCLS * CC)   // 38912
#define OFF_COUNTS  (OFF_SUMS_T + BB * NCLS * CC)   // 77824
#define OFF_MEANS_S (OFF_COUNTS + BB * NCLS)        // 77976
#define OFF_MEANS_T (OFF_MEANS_S + BB * NCLS * CC)  // 116888
#define OFF_NORM_S  (OFF_MEANS_T + BB * NCLS * CC)  // 155800
#define OFF_NORM_T  (OFF_NORM_S + BB * NCLS)        // 155952
#define OFF_LOSS    (OFF_NORM_T + BB * NCLS)        // 156104
#define WS_FLOATS   (OFF_LOSS + 1)                  // 156105 (~610KB)

typedef __attribute__((ext_vector_type(2))) float v2f;
typedef __attribute__((ext_vector_type(8))) float v8f;

// ---------------------------------------------------------------------------
// Kernel 0: zero workspace (must happen every call; harness does not re-init)
// ---------------------------------------------------------------------------
__global__ void zero_ws_kernel(float* __restrict__ ws, int n) {
    int i = blockIdx.x * blockDim.x + threadIdx.x;
    if (i < n) ws[i] = 0.0f;
}

// ---------------------------------------------------------------------------
// Kernel 1: per-(batch,class) channel sums + per-class pixel counts.
// Grid (36 slabs x 8 batches), 256 threads. Each slab = 512 pixels,
// processed as 16 sub-tiles of 32 pixels.
//   - feature tile staged in LDS [256ch][33] (pad 33 -> conflict-free)
//   - per-thread (=per-channel) private accumulators acc[256][21]
//     (stride 21 is odd -> conflict-free indexed read-modify-write)
//   - flush with global fp32 atomics (19 per channel per block)
// ---------------------------------------------------------------------------
__global__ __launch_bounds__(256) void class_sums_kernel(
    const float* __restrict__ feat, const int* __restrict__ target,
    float* __restrict__ sums, float* __restrict__ counts, int doCounts)
{
    __shared__ float tile[256 * 33];   // 33.8 KB
    __shared__ float acc[256 * 21];    // 21.5 KB
    __shared__ int   lbl[32];
    __shared__ int   hist[NCLS];

    const int t    = threadIdx.x;
    const int b    = blockIdx.y;
    const int slab = blockIdx.x;       // 36 slabs * 512 px = 18432
    const int w    = t >> 5;
    const int lane = t & 31;

    #pragma unroll
    for (int k = 0; k < 21; ++k) acc[t * 21 + k] = 0.0f;
    if (t < NCLS) hist[t] = 0;

    const float* fb = feat + (size_t)b * CC * HWP;

    for (int sub = 0; sub < 16; ++sub) {
        const int p0 = slab * 512 + sub * 32;
        __syncthreads();  // previous accumulate done before lbl/tile overwrite
        if (t < 32) {
            int l = target[b * HWP + p0 + t];
            l = (l >= 0 && l < NCLS) ? l : 0;
            lbl[t] = l;
            if (doCounts) atomicAdd(&hist[l], 1);
        }
        // cooperative coalesced load: wave w loads 32 consecutive pixels of
        // channels {i*8 + w}, i = 0..31  (128B per wave per load)
        #pragma unroll 4
        for (int i = 0; i < 32; ++i) {
            const int ch = i * 8 + w;
            tile[ch * 33 + lane] = fb[(size_t)ch * HWP + p0 + lane];
        }
        __syncthreads();
        // thread t owns channel t; scatter-accumulate into private LDS rows
        #pragma unroll 4
        for (int j = 0; j < 32; ++j) {
            const int k = lbl[j];                 // broadcast read
            acc[t * 21 + k] += tile[t * 33 + j];  // conflict-free strides
        }
    }
    __syncthreads();
    #pragma unroll
    for (int k = 0; k < NCLS; ++k)
        atomicAdd(&sums[((size_t)b * NCLS + k) * CC + t], acc[t * 21 + k]);
    if (doCounts && t < NCLS)
        atomicAdd(&counts[b * NCLS + t], (float)hist[t]);
}

// ---------------------------------------------------------------------------
// Kernel 2: means = sums/(count+eps), center norms = sqrt(sum_c mean^2).
// Grid (19 x 8), 256 threads (one per channel).
// ---------------------------------------------------------------------------
__global__ __launch_bounds__(256) void means_norms_kernel(
    const float* __restrict__ sums, const float* __restrict__ counts,
    float* __restrict__ means, float* __restrict__ norms)
{
    __shared__ float red[256];
    const int b = blockIdx.y, k = blockIdx.x, c = threadIdx.x;
    const float cnt = counts[b * NCLS + k];
    const float m = sums[((size_t)b * NCLS + k) * CC + c] / (cnt + EPS_MEAN);
    means[((size_t)b * NCLS + k) * CC + c] = m;
    red[c] = m * m;
    __syncthreads();
    for (int s = 128; s > 0; s >>= 1) {
        if (c < s) red[c] += red[c + s];
        __syncthreads();
    }
    if (c == 0) norms[b * NCLS + k] = sqrtf(red[0]);
}

// ---------------------------------------------------------------------------
// Kernel 3: WMMA pixel-vs-center dots + loss accumulation.
// Grid (144 x 8), 256 threads = 8 waves; each wave owns 16 pixels.
// Per wave: D[16px x 16cls] tiles via V_WMMA_F32_16X16X4_F32, K-dim = channels
// (4 per instruction, 64 steps), two class tiles (19 padded to 32 with a
// zeroed row 19 for class indices >= 19). 256 wmma ops per wave.
//
// A fragment (16x4 f32): lane l -> pixel m=l&15; VGPR v -> channel
//   c0 + v + 2*(l>>4). Lanes 0-15 hit 64B-contiguous pixels (pixel is the
//   fastest-varying memory dim), lanes 16-31 the matching 64B at channel+2.
// B fragment (4x16 f32): lane l -> class n=l&15 (tile-offset), VGPR v ->
//   channel row c0 + v + 2*(l>>4); read from LDS means with row pad 260
//   -> conflict-free (bank = (4k + c) & 63 distinct across half-waves).
// ---------------------------------------------------------------------------
__global__ __launch_bounds__(256) void wmma_loss_kernel(
    const float* __restrict__ featS, const float* __restrict__ featT,
    const int* __restrict__ target,
    const float* __restrict__ meansS, const float* __restrict__ meansT,
    const float* __restrict__ normS, const float* __restrict__ normT,
    float* __restrict__ lossAcc)
{
    const int MPAD = 260;
    __shared__ float mS[20 * 260];     // 20.8 KB (row 19 = zeros)
    __shared__ float mT[20 * 260];     // 20.8 KB
    __shared__ float gst[8 * 16 * 33]; // 16.9 KB D-tile staging (pad 33)
    __shared__ int   lbls[128];

    const int t     = threadIdx.x;
    const int b     = blockIdx.y;
    const int pBase = blockIdx.x * 128;

    // stage means (19 rows, coalesced) + zero the padding row
    for (int k = 0; k < NCLS; ++k) {
        mS[k * MPAD + t] = meansS[((size_t)b * NCLS + k) * CC + t];
        mT[k * MPAD + t] = meansT[((size_t)b * NCLS + k) * CC + t];
    }
    mS[NCLS * MPAD + t] = 0.0f;
    mT[NCLS * MPAD + t] = 0.0f;
    if (t < 128) {
        int l = target[b * HWP + pBase + t];
        lbls[t] = (l >= 0 && l < NCLS) ? l : 0;
    }
    __syncthreads();

    const int wv   = t >> 5;
    const int lane = t & 31;
    const int m    = lane & 15;        // pixel row within wave's 16
    const int half = lane >> 4;
    const int n    = lane & 15;        // class column within tile
    const int p    = pBase + wv * 16 + m;
    const int k0r  = n;                                       // tile 0 row
    const int k1r  = (16 + n < NCLS) ? (16 + n) : NCLS;       // tile 1 row (pad->19)

    const float* fS = featS + (size_t)b * CC * HWP + p;
    const float* fT = featT + (size_t)b * CC * HWP + p;

    v8f accS0 = {}, accS1 = {}, accT0 = {}, accT1 = {};
    float sqS = 0.0f, sqT = 0.0f;

    for (int c0 = 0; c0 < CC; c0 += 4) {
        const int cA = c0 + 2 * half;
        v2f a_s, a_t, bS0, bS1, bT0, bT1;
        a_s.x = fS[(size_t)cA * HWP];
        a_s.y = fS[(size_t)(cA + 1) * HWP];
        a_t.x = fT[(size_t)cA * HWP];
        a_t.y = fT[(size_t)(cA + 1) * HWP];
        sqS += a_s.x * a_s.x + a_s.y * a_s.y;
        sqT += a_t.x * a_t.x + a_t.y * a_t.y;

        bS0.x = mS[k0r * MPAD + cA]; bS0.y = mS[k0r * MPAD + cA + 1];
        bS1.x = mS[k1r * MPAD + cA]; bS1.y = mS[k1r * MPAD + cA + 1];
        bT0.x = mT[k0r * MPAD + cA]; bT0.y = mT[k0r * MPAD + cA + 1];
        bT1.x = mT[k1r * MPAD + cA]; bT1.y = mT[k1r * MPAD + cA + 1];

        accS0 = __builtin_amdgcn_wmma_f32_16x16x4_f32(false, a_s, false, bS0,
                                                      (short)0, accS0, false, false);
        accS1 = __builtin_amdgcn_wmma_f32_16x16x4_f32(false, a_s, false, bS1,
                                                      (short)0, accS1, false, false);
        accT0 = __builtin_amdgcn_wmma_f32_16x16x4_f32(false, a_t, false, bT0,
                                                      (short)0, accT0, false, false);
        accT1 = __builtin_amdgcn_wmma_f32_16x16x4_f32(false, a_t, false, bT1,
                                                      (short)0, accT1, false, false);
    }

    // per-pixel feature norms: halves hold disjoint channel subsets
    const float nxS = sqrtf(sqS + __shfl_xor(sqS, 16, 32));
    const float nxT = sqrtf(sqT + __shfl_xor(sqT, 16, 32));

    float* g = gst + wv * 16 * 33;

    // ---- S: stage D tiles, gather G[m, label(m)] ----
    #pragma unroll
    for (int r = 0; r < 8; ++r) {
        const int mr = r + 8 * half;   // C/D layout: VGPR r -> rows r / r+8
        g[mr * 33 + n]      = accS0[r];
        g[mr * 33 + 16 + n] = accS1[r];
    }
    __syncthreads();
    float simS = 0.0f;
    int kme = 0;
    if (half == 0) {
        kme = lbls[wv * 16 + m];
        const float gv = g[m * 33 + kme];
        const float nc = normS[b * NCLS + kme];
        simS = gv / fmaxf(nxS * nc, EPS_COS);
    }
    __syncthreads();
    // ---- T ----
    #pragma unroll
    for (int r = 0; r < 8; ++r) {
        const int mr = r + 8 * half;
        g[mr * 33 + n]      = accT0[r];
        g[mr * 33 + 16 + n] = accT1[r];
    }
    __syncthreads();
    float val = 0.0f;
    if (half == 0) {
        const float gv = g[m * 33 + kme];
        const float nc = normT[b * NCLS + kme];
        const float simT = gv / fmaxf(nxT * nc, EPS_COS);
        const float d = simS - simT;
        val = d * d * INV_N;
    }
    // wave reduction, one global atomic per wave
    val += __shfl_xor(val, 1, 32);
    val += __shfl_xor(val, 2, 32);
    val += __shfl_xor(val, 4, 32);
    val += __shfl_xor(val, 8, 32);
    val += __shfl_xor(val, 16, 32);
    if (lane == 0) atomicAdd(lossAcc, val);
}

// ---------------------------------------------------------------------------
// Kernel 4: publish scalar
// ---------------------------------------------------------------------------
__global__ void finalize_kernel(const float* __restrict__ lossAcc,
                                float* __restrict__ out) {
    out[0] = lossAcc[0];
}

extern "C" void kernel_launch(void* const* d_in, const int* in_sizes, int n_in,
                              void* d_out, int out_size, void* d_ws, size_t ws_size,
                              hipStream_t stream) {
    const float* featS  = (const float*)d_in[0];
    const float* featT  = (const float*)d_in[1];
    const int*   target = (const int*)d_in[2];
    float* out = (float*)d_out;
    float* ws  = (float*)d_ws;

    float* sumsS  = ws + OFF_SUMS_S;
    float* sumsT  = ws + OFF_SUMS_T;
    float* counts = ws + OFF_COUNTS;
    float* meansS = ws + OFF_MEANS_S;
    float* meansT = ws + OFF_MEANS_T;
    float* normS  = ws + OFF_NORM_S;
    float* normT  = ws + OFF_NORM_T;
    float* loss   = ws + OFF_LOSS;

    zero_ws_kernel<<<(WS_FLOATS + 255) / 256, 256, 0, stream>>>(ws, WS_FLOATS);

    dim3 g1(HWP / 512, BB);  // 36 x 8
    class_sums_kernel<<<g1, 256, 0, stream>>>(featS, target, sumsS, counts, 1);
    class_sums_kernel<<<g1, 256, 0, stream>>>(featT, target, sumsT, counts, 0);

    dim3 g2(NCLS, BB);       // 19 x 8
    means_norms_kernel<<<g2, 256, 0, stream>>>(sumsS, counts, meansS, normS);
    means_norms_kernel<<<g2, 256, 0, stream>>>(sumsT, counts, meansT, normT);

    dim3 g3(HWP / 128, BB);  // 144 x 8
    wmma_loss_kernel<<<g3, 256, 0, stream>>>(featS, featT, target,
                                             meansS, meansT, normS, normT, loss);

    finalize_kernel<<<1, 1, 0, stream>>>(loss, out);
}